// Discriminator_32555852104187
// MI455X (gfx1250) — compile-verified
//
#include <hip/hip_runtime.h>
#include <hip/hip_bf16.h>
#include <math.h>

// MolGAN discriminator, algebraically collapsed (no nonlinearity until the
// aggregation gate). Per batch b:
//   P   = sum_r adj[b,:,:,1:] + I                  (128x128, LDS)
//   U   = P @ [node | 1 | 0..]                     (WMMA f32 16x16x4, K=128)
//   V   = P @ U                                    (WMMA)
//   Z   = [V[:,0:5] | node | V[:,5] | U[:,5] | 1]  (128x16)
//   pre_s/pre_t = Z @ Ms/Mt  (Ms,Mt are 16x128 derived weight blocks) (WMMA)
//   g0  = tanh( sum_i sigmoid(pre_s)*tanh(pre_t) ) ; tiny dense tail in VALU.
// Memory-bound on the adj stream (~168 MB -> ~7 us @ 23.3 TB/s); adj is read
// with aligned b128 loads (5 float4 per 4 matrix elements).

typedef __attribute__((ext_vector_type(2))) float v2f;
typedef __attribute__((ext_vector_type(8))) float v8f;

__device__ __forceinline__ v8f wmma4(v2f a, v2f b, v8f c) {
  // D = A(16x4) x B(4x16) + C(16x16), f32, wave32
  return __builtin_amdgcn_wmma_f32_16x16x4_f32(false, a, false, b, (short)0, c,
                                               false, false);
}

#define PA 129   // pitch of P (coprime with 64 banks for column reads)
#define PZ 17    // pitch of the narrow 128x16 buffers
#define PM 136   // pitch of Ms/Mt rows (keeps the two lane-halves on disjoint banks)

__global__ __launch_bounds__(256) void disc_kernel(
    const float* __restrict__ adj, const float* __restrict__ node,
    const float* __restrict__ W1, const float* __restrict__ b1,
    const float* __restrict__ W2, const float* __restrict__ b2,
    const float* __restrict__ Ws, const float* __restrict__ bs,
    const float* __restrict__ Wt, const float* __restrict__ bt,
    const float* __restrict__ Wl1, const float* __restrict__ bl1,
    const float* __restrict__ Wl2, const float* __restrict__ bl2,
    const float* __restrict__ Wo, const float* __restrict__ bo,
    float* __restrict__ out0, float* __restrict__ outg) {
  __shared__ float sP[128 * PA];   // P = A + I
  __shared__ float sN[128 * PZ];   // node augmented [node | 1 | 0]
  __shared__ float sU[128 * PZ];   // P @ naug
  __shared__ float sV[128 * PZ];   // P @ U
  __shared__ float sZ[128 * PZ];   // aggregation factor
  __shared__ float sMs[16 * PM];   // derived gate weights
  __shared__ float sMt[16 * PM];
  __shared__ float sW12[5 * 64];   // W1 @ W2
  __shared__ float sbW[64];        // b1 @ W2
  __shared__ float sG0[128];
  __shared__ float sG1[128];
  __shared__ float sG2[64];

  const int b = blockIdx.x;
  const int tid = threadIdx.x;
  const int lane = tid & 31;
  const int wv = tid >> 5;   // wave 0..7
  const int l15 = lane & 15; // row/col within tile
  const int lh = lane >> 4;  // lane half (K/M split per ISA layout)

  // ---- Phase 0a: P = sum_{r=1..4} adj[b,i,j,r] + I ----------------------
  // Each group of 4 consecutive (i,j) elements spans 20 floats = 5 aligned
  // float4 -> pure global_load_b128 streaming of the 320KB adj[b] slab.
  {
    const float4* adj4 = (const float4*)(adj + (size_t)b * (128 * 128 * 5));
    for (int g = tid; g < 128 * 128 / 4; g += 256) {
      const float4* p = adj4 + g * 5;
      float4 f0 = p[0], f1 = p[1], f2 = p[2], f3 = p[3], f4 = p[4];
      float s0 = f0.y + f0.z + f0.w + f1.x;   // elem 0: flats 1..4
      float s1 = f1.z + f1.w + f2.x + f2.y;   // elem 1: flats 6..9
      float s2 = f2.w + f3.x + f3.y + f3.z;   // elem 2: flats 11..14
      float s3 = f4.x + f4.y + f4.z + f4.w;   // elem 3: flats 16..19
      int e = g * 4;
      int i = e >> 7, j = e & 127;            // 4 consecutive j in row i
      float* dst = &sP[i * PA + j];
      dst[0] = s0 + ((i == j + 0) ? 1.0f : 0.0f);
      dst[1] = s1 + ((i == j + 1) ? 1.0f : 0.0f);
      dst[2] = s2 + ((i == j + 2) ? 1.0f : 0.0f);
      dst[3] = s3 + ((i == j + 3) ? 1.0f : 0.0f);
    }
  }
  // ---- Phase 0b: augmented node matrix [node | ones | zeros] ----
  const float* nodeB = node + (size_t)b * (128 * 5);
  for (int idx = tid; idx < 128 * 16; idx += 256) {
    int i = idx >> 4, c = idx & 15;
    float v = 0.0f;
    if (c < 5) v = nodeB[i * 5 + c];
    else if (c == 5) v = 1.0f;
    sN[i * PZ + c] = v;
  }
  // ---- Phase 0c: W12 = W1@W2 (5x64), bW = b1@W2 (64) ----
  for (int idx = tid; idx < 384; idx += 256) {
    if (idx < 320) {
      int d = idx >> 6, k = idx & 63;
      float s = 0.0f;
      for (int j = 0; j < 128; ++j) s += W1[d * 128 + j] * W2[j * 64 + k];
      sW12[d * 64 + k] = s;
    } else {
      int k = idx - 320;
      float s = 0.0f;
      for (int j = 0; j < 128; ++j) s += b1[j] * W2[j * 64 + k];
      sbW[k] = s;
    }
  }
  __syncthreads();

  // ---- Phase 0d: Ms/Mt [16 x 128] so that pre_s = Z @ Ms, pre_t = Z @ Mt ----
  // rows 0-4 : W12 @ Wx[0:64]   (h2-linear part via Q = P^2 @ node)
  // rows 5-9 : Wx[64:69]        (direct node part of ann)
  // row 10   : bW  @ Wx[0:64]   (coefficient of r2 = P^2 @ 1)
  // row 11   : b2  @ Wx[0:64]   (coefficient of r1 = P @ 1)
  // row 12   : bias bx          (coefficient of the ones column)
  {
    int a = tid & 127;
    const float* Wx = (tid < 128) ? Ws : Wt;
    const float* bx = (tid < 128) ? bs : bt;
    float* M = (tid < 128) ? sMs : sMt;
    float r0 = 0, r1 = 0, r2 = 0, r3 = 0, r4 = 0, r10 = 0, r11 = 0;
    for (int k = 0; k < 64; ++k) {
      float w = Wx[k * 128 + a];
      r0 += sW12[0 * 64 + k] * w;
      r1 += sW12[1 * 64 + k] * w;
      r2 += sW12[2 * 64 + k] * w;
      r3 += sW12[3 * 64 + k] * w;
      r4 += sW12[4 * 64 + k] * w;
      r10 += sbW[k] * w;
      r11 += b2[k] * w;
    }
    M[0 * PM + a] = r0; M[1 * PM + a] = r1; M[2 * PM + a] = r2;
    M[3 * PM + a] = r3; M[4 * PM + a] = r4;
    for (int d = 0; d < 5; ++d) M[(5 + d) * PM + a] = Wx[(64 + d) * 128 + a];
    M[10 * PM + a] = r10;
    M[11 * PM + a] = r11;
    M[12 * PM + a] = bx[a];
    M[13 * PM + a] = 0.0f; M[14 * PM + a] = 0.0f; M[15 * PM + a] = 0.0f;
  }
  __syncthreads();

  // ---- GEMM1: U = P @ naug  (one 16x16 tile per wave, K=128) ----
  {
    const int rbase = wv * 16 + l15;
    v8f acc = {0.f, 0.f, 0.f, 0.f, 0.f, 0.f, 0.f, 0.f};
#pragma unroll 8
    for (int k0 = 0; k0 < 128; k0 += 4) {
      v2f a, bb;
      a.x = sP[rbase * PA + k0 + 2 * lh];
      a.y = sP[rbase * PA + k0 + 2 * lh + 1];
      bb.x = sN[(k0 + 2 * lh) * PZ + l15];
      bb.y = sN[(k0 + 2 * lh + 1) * PZ + l15];
      acc = wmma4(a, bb, acc);
    }
#pragma unroll
    for (int v = 0; v < 8; ++v) sU[(wv * 16 + v + 8 * lh) * PZ + l15] = acc[v];
  }
  __syncthreads();

  // ---- GEMM2: V = P @ U ----
  {
    const int rbase = wv * 16 + l15;
    v8f acc = {0.f, 0.f, 0.f, 0.f, 0.f, 0.f, 0.f, 0.f};
#pragma unroll 8
    for (int k0 = 0; k0 < 128; k0 += 4) {
      v2f a, bb;
      a.x = sP[rbase * PA + k0 + 2 * lh];
      a.y = sP[rbase * PA + k0 + 2 * lh + 1];
      bb.x = sU[(k0 + 2 * lh) * PZ + l15];
      bb.y = sU[(k0 + 2 * lh + 1) * PZ + l15];
      acc = wmma4(a, bb, acc);
    }
#pragma unroll
    for (int v = 0; v < 8; ++v) sV[(wv * 16 + v + 8 * lh) * PZ + l15] = acc[v];
  }
  __syncthreads();

  // ---- Build Z = [Q | node | r2 | r1 | 1 | 0 0 0] ----
  for (int idx = tid; idx < 128 * 16; idx += 256) {
    int i = idx >> 4, c = idx & 15;
    float v;
    if (c < 5) v = sV[i * PZ + c];              // Q = P^2 @ node
    else if (c < 10) v = sN[i * PZ + (c - 5)];  // node
    else if (c == 10) v = sV[i * PZ + 5];       // r2
    else if (c == 11) v = sU[i * PZ + 5];       // r1
    else if (c == 12) v = 1.0f;                 // bias lane
    else v = 0.0f;
    sZ[i * PZ + c] = v;
  }
  __syncthreads();

  // ---- GEMM3 + fused gate/pool: each wave owns a 16-wide column strip ----
  {
    const int cb = wv * 16;       // strip base column of AUX
    float pAcc = 0.0f;            // per-lane pool over this lane's rows
    for (int ti = 0; ti < 8; ++ti) {
      const int rbase = ti * 16 + l15;
      v8f sacc = {0.f, 0.f, 0.f, 0.f, 0.f, 0.f, 0.f, 0.f};
      v8f tacc = {0.f, 0.f, 0.f, 0.f, 0.f, 0.f, 0.f, 0.f};
#pragma unroll
      for (int k0 = 0; k0 < 16; k0 += 4) {
        v2f a, bsf, btf;
        a.x = sZ[rbase * PZ + k0 + 2 * lh];
        a.y = sZ[rbase * PZ + k0 + 2 * lh + 1];
        bsf.x = sMs[(k0 + 2 * lh) * PM + cb + l15];
        bsf.y = sMs[(k0 + 2 * lh + 1) * PM + cb + l15];
        btf.x = sMt[(k0 + 2 * lh) * PM + cb + l15];
        btf.y = sMt[(k0 + 2 * lh + 1) * PM + cb + l15];
        sacc = wmma4(a, bsf, sacc);
        tacc = wmma4(a, btf, tacc);
      }
#pragma unroll
      for (int v = 0; v < 8; ++v) {
        float gate = 1.0f / (1.0f + __expf(-sacc[v]));
        pAcc += gate * tanhf(tacc[v]);
      }
    }
    pAcc += __shfl_xor(pAcc, 16, 32);  // combine the two lane-halves (row sets)
    if (lh == 0) sG0[cb + l15] = tanhf(pAcc);
  }
  __syncthreads();

  // ---- Dense tail (tiny, VALU) ----
  if (tid < 128) {
    float s = bl1[tid];
    for (int k = 0; k < 128; ++k) s += sG0[k] * Wl1[k * 128 + tid];
    sG1[tid] = s;
  }
  __syncthreads();
  if (tid < 64) {
    float s = bl2[tid];
    for (int k = 0; k < 128; ++k) s += sG1[k] * Wl2[k * 64 + tid];
    sG2[tid] = s;
    outg[b * 64 + tid] = s;  // second tuple output: g [B,64]
  }
  __syncthreads();
  if (tid == 0) {
    float s = bo[0];
    for (int k = 0; k < 64; ++k) s += sG2[k] * Wo[k];
    out0[b] = s;  // first tuple output: out [B,1]
  }
}

extern "C" void kernel_launch(void* const* d_in, const int* in_sizes, int n_in,
                              void* d_out, int out_size, void* d_ws,
                              size_t ws_size, hipStream_t stream) {
  const float* adj = (const float*)d_in[0];
  const float* node = (const float*)d_in[1];
  const float* W1 = (const float*)d_in[2];
  const float* b1 = (const float*)d_in[3];
  const float* W2 = (const float*)d_in[4];
  const float* b2 = (const float*)d_in[5];
  const float* Ws = (const float*)d_in[6];
  const float* bs = (const float*)d_in[7];
  const float* Wt = (const float*)d_in[8];
  const float* bt = (const float*)d_in[9];
  const float* Wl1 = (const float*)d_in[10];
  const float* bl1 = (const float*)d_in[11];
  const float* Wl2 = (const float*)d_in[12];
  const float* bl2 = (const float*)d_in[13];
  const float* Wo = (const float*)d_in[14];
  const float* bo = (const float*)d_in[15];

  float* out0 = (float*)d_out;        // [512] = out flattened
  float* outg = (float*)d_out + 512;  // [512*64] = g flattened

  disc_kernel<<<512, 256, 0, stream>>>(adj, node, W1, b1, W2, b2, Ws, bs, Wt,
                                       bt, Wl1, bl1, Wl2, bl2, Wo, bo, out0,
                                       outg);
}